// U_Social_Aggregator_13168369729718
// MI455X (gfx1250) — compile-verified
//
#include <hip/hip_runtime.h>
#include <hip/hip_bf16.h>
#include <stdint.h>

// Segment-mean gather:  out[n,:] = mean_{e in seg n} u2e[neigh_idx[e], :]
//
// Memory-bound (AI ~ 0.24 FLOP/B, roofline ~37us at 23.3 TB/s): the whole
// game is the random 256B-row gather. CDNA5 path: double-buffered
// global_load_async_to_lds_b128 pipeline (ASYNCcnt, in-order completion ->
// constant-immediate s_wait_asynccnt 0x8 overlaps gather with accumulation),
// consumed via ds_load_b64.

#define EMBED            64
#define BATCH            16            // rows per pipeline stage (4 KB)
#define WAVES_PER_BLOCK  8
#define BLOCK_THREADS    (WAVES_PER_BLOCK * 32)

// ---------------------------------------------------------------------------
// Kernel 1: segment_ids (sorted) -> CSR starts[N+1] in workspace.
// Every entry of starts[0..N] is written exactly once per call (covers empty
// segments, leading and trailing ranges) -> deterministic, poison-safe.
// ---------------------------------------------------------------------------
__global__ void seg_starts_kernel(const int* __restrict__ seg,
                                  int* __restrict__ starts,
                                  int E, int N) {
    int e = blockIdx.x * blockDim.x + threadIdx.x;
    if (e >= E) return;
    int s  = seg[e];
    int sp = (e == 0) ? -1 : seg[e - 1];
    if (s != sp) {
        for (int t = sp + 1; t <= s; ++t) starts[t] = e;   // boundary fill
    }
    if (e == E - 1) {
        for (int t = s + 1; t <= N; ++t) starts[t] = E;    // trailing fill
    }
}

// Issue one full stage: 8x global_load_async_to_lds_b128.
//   lanes 0-15  fetch row (2p)   : 16B/lane -> 256B row
//   lanes 16-31 fetch row (2p+1) : 16B/lane -> 256B row
// Padded slots (>= cnt) are clamped to a valid edge; never accumulated.
__device__ __forceinline__ void issue_stage(const float* __restrict__ u2e,
                                            const int* __restrict__ nidx,
                                            int base, int cnt, uint32_t dst_lds,
                                            int lane, int rsel, int sub) {
    const int e   = base + (lane < cnt ? lane : 0);   // lane l: edge base+l (clamped)
    const int row = nidx[e];
#pragma unroll
    for (int p = 0; p < BATCH / 2; ++p) {
        const int src  = 2 * p + rsel;
        const int ridx = __shfl(row, src, 32);        // uniform src -> broadcast
        const float*   gp = u2e + (size_t)ridx * EMBED + sub * 4;       // 16B chunk
        const uint32_t d  = dst_lds + (uint32_t)(p * 512 + lane * 16);
        asm volatile("global_load_async_to_lds_b128 %0, %1, off"
                     :: "v"(d), "v"(gp)
                     : "memory");
    }
}

// ---------------------------------------------------------------------------
// Kernel 2: one wave32 per output node, double-buffered async gather.
// ---------------------------------------------------------------------------
__global__ void __launch_bounds__(BLOCK_THREADS)
seg_mean_kernel(const float* __restrict__ u2e,
                const int* __restrict__ nidx,
                const int* __restrict__ starts,
                float* __restrict__ out, int N) {
    __shared__ __align__(16) float stage[WAVES_PER_BLOCK * 2 * BATCH * EMBED]; // 64 KB

    const int lane = threadIdx.x & 31;
    const int wave = threadIdx.x >> 5;
    const int node = blockIdx.x * WAVES_PER_BLOCK + wave;
    if (node >= N) return;                     // whole wave exits; no barriers used

    const int start = starts[node];
    const int end   = starts[node + 1];

    float* buf0 = &stage[(wave * 2 + 0) * BATCH * EMBED];
    float* buf1 = &stage[(wave * 2 + 1) * BATCH * EMBED];
    const uint32_t lds0 = (uint32_t)(uintptr_t)buf0;  // AS(3) byte offsets
    const uint32_t lds1 = (uint32_t)(uintptr_t)buf1;
    const int rsel = lane >> 4;                // 0: even row of pair, 1: odd row
    const int sub  = lane & 15;                // 16B chunk within the 256B row

    float accx = 0.0f, accy = 0.0f;

    const int total  = end - start;
    const int nbatch = (total + BATCH - 1) / BATCH;

    if (nbatch > 0) {
        // prologue: stage 0 -> buf0
        issue_stage(u2e, nidx, start, total < BATCH ? total : BATCH,
                    lds0, lane, rsel, sub);

        for (int b = 0; b < nbatch; ++b) {
            const int base = start + b * BATCH;
            const int rem  = end - base;
            const int cnt  = rem < BATCH ? rem : BATCH;

            if (b + 1 < nbatch) {
                const int nbase = base + BATCH;
                const int nrem  = end - nbase;
                const int ncnt  = nrem < BATCH ? nrem : BATCH;
                // buffer being refilled was last ds-read in batch b-1:
                // make sure those reads retired before the async engine writes it
                asm volatile("s_wait_dscnt 0x0" ::: "memory");
                issue_stage(u2e, nidx, nbase, ncnt,
                            ((b + 1) & 1) ? lds1 : lds0, lane, rsel, sub);
                // async loads complete in order: <=8 outstanding means the
                // previous stage's 8 transfers (batch b) have landed
                asm volatile("s_wait_asynccnt 0x8" ::: "memory");
            } else {
                asm volatile("s_wait_asynccnt 0x0" ::: "memory");
            }

            // each lane owns columns {2*lane, 2*lane+1}; sum the staged rows
            const float* stg = (b & 1) ? buf1 : buf0;
            for (int r = 0; r < cnt; ++r) {
                const float2 v = *(const float2*)(stg + r * EMBED + lane * 2);
                accx += v.x;
                accy += v.y;
            }
        }
    }

    const float inv = 1.0f / (float)(total > 0 ? total : 1);  // matches max(counts,1)
    float2 res;
    res.x = accx * inv;
    res.y = accy * inv;
    *(float2*)(out + (size_t)node * EMBED + lane * 2) = res;  // 256B coalesced store
}

// ---------------------------------------------------------------------------
// Launch: inputs (u2e_weight f32 [1e6*64], neigh_idx i32 [E], segment_ids
// i32 [E]); output f32 [N*64], N = out_size / 64. d_ws holds starts[N+1].
// ---------------------------------------------------------------------------
extern "C" void kernel_launch(void* const* d_in, const int* in_sizes, int n_in,
                              void* d_out, int out_size, void* d_ws, size_t ws_size,
                              hipStream_t stream) {
    const float* u2e  = (const float*)d_in[0];
    const int*   nidx = (const int*)d_in[1];
    const int*   seg  = (const int*)d_in[2];
    float*       out  = (float*)d_out;

    const int E = in_sizes[1];
    const int N = out_size / EMBED;

    int* starts = (int*)d_ws;                    // (N+1) ints

    seg_starts_kernel<<<(E + 255) / 256, 256, 0, stream>>>(seg, starts, E, N);

    const int blocks = (N + WAVES_PER_BLOCK - 1) / WAVES_PER_BLOCK;
    seg_mean_kernel<<<blocks, BLOCK_THREADS, 0, stream>>>(u2e, nidx, starts, out, N);
}